// RangePredictor_56075093016590
// MI455X (gfx1250) — compile-verified
//
#include <hip/hip_runtime.h>
#include <math.h>

typedef __attribute__((ext_vector_type(2))) float v2f;
typedef __attribute__((ext_vector_type(8))) float v8f;

#define N_NODES 100000
#define N_EDGES 1200000

// ---------------- init / utility ----------------

__global__ void init_kernel(float* __restrict__ deg, float* __restrict__ acc) {
    int i = blockIdx.x * blockDim.x + threadIdx.x;
    if (i < N_NODES) deg[i] = 1.0f;          // self-loop contribution
    if (i < 32)      acc[i] = 0.0f;          // reduction accumulators
}

__global__ void zero_kernel(float* __restrict__ p, long n) {
    long i = (long)blockIdx.x * blockDim.x + threadIdx.x;
    if (i < n) p[i] = 0.0f;
}

__global__ void deg_kernel(const int* __restrict__ dst, float* __restrict__ deg) {
    int e = blockIdx.x * blockDim.x + threadIdx.x;
    if (e < N_EDGES) atomicAdd(&deg[dst[e]], 1.0f);
}

__global__ void rsqrt_kernel(float* __restrict__ deg) {
    int i = blockIdx.x * blockDim.x + threadIdx.x;
    if (i < N_NODES) deg[i] = rsqrtf(deg[i]);
}

// ---------------- layer-1 GEMM: x(N x 5) @ W1(5 x 64) ----------------

__global__ void gemm_in_kernel(const float* __restrict__ x, const float* __restrict__ W,
                               float* __restrict__ out) {
    int t = blockIdx.x * blockDim.x + threadIdx.x;
    if (t >= N_NODES * 64) return;
    int i = t >> 6, c = t & 63;
    const float* xr = x + i * 5;
    float s = xr[0] * W[c] + xr[1] * W[64 + c] + xr[2] * W[128 + c] +
              xr[3] * W[192 + c] + xr[4] * W[256 + c];
    out[t] = s;
}

// ---------------- f32 WMMA GEMM: in(N x K) @ W(K x O) ----------------
// One wave32 computes one 16x16 output tile via V_WMMA_F32_16X16X4_F32.
// A frag (16x4): lane<16 -> K=k,k+1 of row (lane&15); lane>=16 -> K=k+2,k+3.
// B frag (4x16): same split over K, N = lane&15.
// D (16x16): VGPR v holds row v (lanes 0-15) / row v+8 (lanes 16-31).
template<int KDIM, int ODIM>
__global__ void wmma_gemm_kernel(const float* __restrict__ in, const float* __restrict__ W,
                                 float* __restrict__ out, int ntiles_m) {
    const int ntn  = ODIM / 16;
    int wave = (int)((blockIdx.x * blockDim.x + threadIdx.x) >> 5);
    if (wave >= ntiles_m * ntn) return;            // wave-uniform: EXEC stays all-ones
    int tm = wave / ntn, tn = wave % ntn;
    int lane = threadIdx.x & 31;
    int half = lane >> 4;
    int r    = lane & 15;

    const float* arow = in + (long)(tm * 16 + r) * KDIM;
    v8f acc = {};
#pragma unroll
    for (int kk = 0; kk < KDIM; kk += 4) {
        v2f a, b;
        int ka = kk + 2 * half;
        a.x = arow[ka];
        a.y = arow[ka + 1];
        b.x = W[(long)ka * ODIM + tn * 16 + r];
        b.y = W[(long)(ka + 1) * ODIM + tn * 16 + r];
        acc = __builtin_amdgcn_wmma_f32_16x16x4_f32(
            /*neg_a=*/false, a, /*neg_b=*/false, b,
            /*c_mod=*/(short)0, acc, /*reuse_a=*/false, /*reuse_b=*/false);
    }
    float* orow = out + (long)(tm * 16 + 8 * half) * ODIM + tn * 16 + r;
#pragma unroll
    for (int v = 0; v < 8; v++) orow[(long)v * ODIM] = acc[v];
}

// ---------------- edge scatter: agg[dst] += h[src] * dinv[src]*dinv[dst] ----------------
template<int F>
__global__ void scatter_kernel(const int* __restrict__ src, const int* __restrict__ dst,
                               const float* __restrict__ dinv, const float* __restrict__ h,
                               float* __restrict__ agg) {
    const int Q = F / 4;                        // float4 chunks per edge
    long t = (long)blockIdx.x * blockDim.x + threadIdx.x;
    if (t >= (long)N_EDGES * Q) return;
    int e = (int)(t / Q);
    int q = (int)(t % Q);
    int s = src[e], d = dst[e];
    float nrm = dinv[s] * dinv[d];
    const float4 hv = *(const float4*)(h + (long)s * F + q * 4);
    float* ap = agg + (long)d * F + q * 4;
    atomicAdd(ap + 0, hv.x * nrm);
    atomicAdd(ap + 1, hv.y * nrm);
    atomicAdd(ap + 2, hv.z * nrm);
    atomicAdd(ap + 3, hv.w * nrm);
}

// ---------------- self-loop + bias (+ReLU): agg = agg + h*dinv^2 + b ----------------
template<int F, bool RELU>
__global__ void finish_kernel(float* __restrict__ agg, const float* __restrict__ h,
                              const float* __restrict__ dinv, const float* __restrict__ b) {
    long t = (long)blockIdx.x * blockDim.x + threadIdx.x;
    if (t >= (long)N_NODES * F) return;
    int i = (int)(t / F), c = (int)(t % F);
    float di = dinv[i];
    float v = agg[t] + h[t] * di * di + b[c];
    if (RELU) v = fmaxf(v, 0.0f);
    agg[t] = v;
}

// ---------------- reductions ----------------

__global__ void reduce_h3_kernel(const float* __restrict__ h3, float* __restrict__ acc) {
    int t = blockIdx.x * blockDim.x + threadIdx.x;
    int c = t & 15;
    int i = t >> 4;
    int stride = (gridDim.x * blockDim.x) >> 4;
    float s = 0.0f;
    for (; i < N_NODES; i += stride) s += h3[(long)i * 16 + c];
    atomicAdd(&acc[c], s);
}

__global__ void reduce_gf_kernel(const float* __restrict__ x, float* __restrict__ acc) {
    int t = blockIdx.x * blockDim.x + threadIdx.x;
    int stride = gridDim.x * blockDim.x;
    float sc = 0, sa = 0, so = 0, sl = 0, sm = 0;
    for (int i = t; i < N_NODES; i += stride) {
        const float* xr = x + (long)i * 5;
        float comp = xr[2];
        sc += comp; sa += xr[3]; so += xr[4];
        sl += xr[0] * comp; sm += xr[1] * comp;
    }
    atomicAdd(&acc[16], sc);
    atomicAdd(&acc[17], sa);
    atomicAdd(&acc[18], so);
    atomicAdd(&acc[19], sl);
    atomicAdd(&acc[20], sm);
}

// ---------------- head: global features + tiny MLP + softplus ----------------

__global__ void head_kernel(const float* __restrict__ acc,
                            const float* __restrict__ T, const float* __restrict__ Tmax,
                            const float* __restrict__ P1, const float* __restrict__ pb1,
                            const float* __restrict__ P2, const float* __restrict__ pb2,
                            const float* __restrict__ P3, const float* __restrict__ pb3,
                            float* __restrict__ out) {
    __shared__ float emb[24], m1[64], m2[32], val[4];
    int j = threadIdx.x;
    if (j < 16) emb[j] = acc[j] / (float)N_NODES;
    if (j == 16) {
        float n_comp = acc[16], n_AND = acc[17], n_OR = acc[18];
        float Tn = T[0] / Tmax[0];
        float safe = fmaxf(n_comp, 1.0f);
        bool  has  = n_comp > 0.0f;
        emb[16] = n_comp;
        emb[17] = n_AND;
        emb[18] = n_OR;
        emb[19] = n_AND + n_OR;
        emb[20] = has ? acc[19] / safe : 0.0f;
        emb[21] = has ? acc[20] / safe : 0.0f;
        emb[22] = Tn * 50.0f;
        emb[23] = (1.0f / (1.0f + Tn)) * 50.0f;
    }
    __syncthreads();
    if (j < 24) out[4 + j] = emb[j];
    // P1: 24x64
    {
        float s = pb1[j];
        for (int k = 0; k < 24; k++) s += emb[k] * P1[k * 64 + j];
        m1[j] = fmaxf(s, 0.0f);
    }
    __syncthreads();
    if (j < 32) {
        float s = pb2[j];
        for (int k = 0; k < 64; k++) s += m1[k] * P2[k * 32 + j];
        m2[j] = fmaxf(s, 0.0f);
    }
    __syncthreads();
    if (j < 4) {
        float s = pb3[j];
        for (int k = 0; k < 32; k++) s += m2[k] * P3[k * 4 + j];
        float xv = s + 2.0f;
        val[j] = (xv > 20.0f) ? xv : log1pf(expf(xv));   // softplus
    }
    __syncthreads();
    if (j == 0) {
        float amin = 1.0f + val[0];
        float amax = amin + val[1] + 0.5f;
        float bmin = 1.0f + val[2];
        float bmax = bmin + val[3] + 0.5f;
        out[0] = amin; out[1] = amax; out[2] = bmin; out[3] = bmax;
    }
}

// ---------------- launch ----------------

extern "C" void kernel_launch(void* const* d_in, const int* in_sizes, int n_in,
                              void* d_out, int out_size, void* d_ws, size_t ws_size,
                              hipStream_t stream) {
    const float* x    = (const float*)d_in[0];
    const int*   ei   = (const int*)d_in[1];
    const float* T    = (const float*)d_in[2];
    const float* Tmax = (const float*)d_in[3];
    const float* W1   = (const float*)d_in[4];
    const float* b1   = (const float*)d_in[5];
    const float* W2   = (const float*)d_in[6];
    const float* b2   = (const float*)d_in[7];
    const float* W3   = (const float*)d_in[8];
    const float* b3   = (const float*)d_in[9];
    const float* P1   = (const float*)d_in[10];
    const float* pb1  = (const float*)d_in[11];
    const float* P2   = (const float*)d_in[12];
    const float* pb2  = (const float*)d_in[13];
    const float* P3   = (const float*)d_in[14];
    const float* pb3  = (const float*)d_in[15];
    float* out = (float*)d_out;

    const int* src = ei;
    const int* dst = ei + N_EDGES;

    // workspace layout (floats): deg[N] | X0[N*64] | X1[N*64] | X2[N*16] | acc[32]
    float* ws  = (float*)d_ws;
    float* deg = ws;
    float* X0  = deg + N_NODES;
    float* X1  = X0 + (long)N_NODES * 64;
    float* X2  = X1 + (long)N_NODES * 64;
    float* acc = X2 + (long)N_NODES * 16;

    const int TB = 256;
    const long n64 = (long)N_NODES * 64;
    const long n16 = (long)N_NODES * 16;

    // degree / normalization
    init_kernel<<<(N_NODES + TB - 1) / TB, TB, 0, stream>>>(deg, acc);
    deg_kernel<<<(N_EDGES + TB - 1) / TB, TB, 0, stream>>>(dst, deg);
    rsqrt_kernel<<<(N_NODES + TB - 1) / TB, TB, 0, stream>>>(deg);   // deg now holds dinv

    // layer 1: t1 = x@W1 ; h1 = relu(agg(t1) + t1*dinv^2 + b1)
    gemm_in_kernel<<<(int)((n64 + TB - 1) / TB), TB, 0, stream>>>(x, W1, X0);
    zero_kernel<<<(int)((n64 + TB - 1) / TB), TB, 0, stream>>>(X1, n64);
    scatter_kernel<64><<<(int)(((long)N_EDGES * 16 + TB - 1) / TB), TB, 0, stream>>>(src, dst, deg, X0, X1);
    finish_kernel<64, true><<<(int)((n64 + TB - 1) / TB), TB, 0, stream>>>(X1, X0, deg, b1);

    // layer 2: t2 = h1@W2 (WMMA f32) ; h2 = relu(agg(t2) + t2*dinv^2 + b2)
    {
        int ntm = N_NODES / 16;                    // 6250
        int waves = ntm * (64 / 16);               // 25000
        wmma_gemm_kernel<64, 64><<<(waves * 32 + TB - 1) / TB, TB, 0, stream>>>(X1, W2, X0, ntm);
    }
    zero_kernel<<<(int)((n64 + TB - 1) / TB), TB, 0, stream>>>(X1, n64);
    scatter_kernel<64><<<(int)(((long)N_EDGES * 16 + TB - 1) / TB), TB, 0, stream>>>(src, dst, deg, X0, X1);
    finish_kernel<64, true><<<(int)((n64 + TB - 1) / TB), TB, 0, stream>>>(X1, X0, deg, b2);

    // layer 3: t3 = h2@W3 (WMMA f32) ; h3 = agg(t3) + t3*dinv^2 + b3
    {
        int ntm = N_NODES / 16;
        int waves = ntm * (16 / 16);               // 6250
        wmma_gemm_kernel<64, 16><<<(waves * 32 + TB - 1) / TB, TB, 0, stream>>>(X1, W3, X0, ntm);
    }
    zero_kernel<<<(int)((n16 + TB - 1) / TB), TB, 0, stream>>>(X2, n16);
    scatter_kernel<16><<<(int)(((long)N_EDGES * 4 + TB - 1) / TB), TB, 0, stream>>>(src, dst, deg, X0, X2);
    finish_kernel<16, false><<<(int)((n16 + TB - 1) / TB), TB, 0, stream>>>(X2, X0, deg, b3);

    // reductions + head
    reduce_h3_kernel<<<256, TB, 0, stream>>>(X2, acc);
    reduce_gf_kernel<<<256, TB, 0, stream>>>(x, acc);
    head_kernel<<<1, 64, 0, stream>>>(acc, T, Tmax, P1, pb1, P2, pb2, P3, pb3, out);
}